// NonLocalBlock_59399397703753
// MI455X (gfx1250) — compile-verified
//
#include <hip/hip_runtime.h>

// ---------------------------------------------------------------------------
// NonLocalBlock (self-attention) for MI455X / gfx1250, full-fp32 WMMA path.
// B=4, H=W=64 (N=4096), C=256, D=32.
// Attention: flash-style online softmax, 64-col j-blocks, 4-way j-split with
// in-block LDS combine (flash-decode) for 4096-wave occupancy.
// ---------------------------------------------------------------------------

typedef float v2f __attribute__((ext_vector_type(2)));
typedef float v8f __attribute__((ext_vector_type(8)));

#define WMMA_F32(A, B, C) \
  __builtin_amdgcn_wmma_f32_16x16x4_f32(false, (A), false, (B), (short)0, (C), false, false)

static __device__ __forceinline__ v2f ld2(const float* p) { return *(const v2f*)p; }
static __device__ __forceinline__ v2f mk2(float a, float b) { v2f r; r.x = a; r.y = b; return r; }

// ---------------------------------------------------------------------------
// Kernel 1: q = x@Wf, k = x@Wg, v = x@Wh   (x:[16384,256], W:[256,32])
// One wave -> 16 rows x 32 cols of all three outputs. 8 waves/block = 128 rows.
// Weights staged in LDS in K-paired float2 layout (64-wide K chunks, 24 KB).
// ---------------------------------------------------------------------------
__global__ __launch_bounds__(256) void qkv_kernel(
    const float* __restrict__ x,
    const float* __restrict__ Wf, const float* __restrict__ Wg, const float* __restrict__ Wh,
    float* __restrict__ q, float* __restrict__ k, float* __restrict__ v) {
  __shared__ v2f wl[3][32 * 32];  // [weight][kk2*32 + n] = {W[2*kk2+c0, n], W[2*kk2+1+c0, n]}
  const int tid = threadIdx.x;
  const int wave = tid >> 5, lane = tid & 31;
  const int m = lane & 15, half = lane >> 4, kOff = half * 2;
  const int row0 = blockIdx.x * 128 + wave * 16;

  v8f aq0 = {}, aq1 = {}, ak0 = {}, ak1 = {}, av0 = {}, av1 = {};

  for (int c0 = 0; c0 < 256; c0 += 64) {
    for (int idx = tid; idx < 32 * 32; idx += 256) {
      const int kk2 = idx >> 5, n = idx & 31;
      const int g = (c0 + kk2 * 2) * 32 + n;
      wl[0][idx] = mk2(Wf[g], Wf[g + 32]);
      wl[1][idx] = mk2(Wg[g], Wg[g + 32]);
      wl[2][idx] = mk2(Wh[g], Wh[g + 32]);
    }
    __syncthreads();
#pragma unroll
    for (int kk = 0; kk < 64; kk += 4) {
      const v2f a = ld2(&x[(row0 + m) * 256 + c0 + kk + kOff]);  // A tile 16x4
      const int bi = ((kk + kOff) >> 1) * 32;
      const v2f bf0 = wl[0][bi + m], bf1 = wl[0][bi + 16 + m];
      const v2f bg0 = wl[1][bi + m], bg1 = wl[1][bi + 16 + m];
      const v2f bh0 = wl[2][bi + m], bh1 = wl[2][bi + 16 + m];
      aq0 = WMMA_F32(a, bf0, aq0); aq1 = WMMA_F32(a, bf1, aq1);
      ak0 = WMMA_F32(a, bg0, ak0); ak1 = WMMA_F32(a, bg1, ak1);
      av0 = WMMA_F32(a, bh0, av0); av1 = WMMA_F32(a, bh1, av1);
    }
    __syncthreads();
  }
#pragma unroll
  for (int vi = 0; vi < 8; ++vi) {
    const int r = (row0 + vi + half * 8) * 32;
    q[r + m] = aq0[vi]; q[r + 16 + m] = aq1[vi];
    k[r + m] = ak0[vi]; k[r + 16 + m] = ak1[vi];
    v[r + m] = av0[vi]; v[r + 16 + m] = av1[vi];
  }
}

// ---------------------------------------------------------------------------
// Kernel 2: flash attention. o = softmax(k . q^T, axis=-1) @ v  (per batch)
// "queries" = k rows, "keys" = q rows, "values" = v rows (matches einsum axes).
// Block = 8 waves = 2 row-tiles x 4 column-splits (1024 cols each).
// Each wave streams 64-col j-blocks (64 WMMAs per block, one butterfly
// reduction set per 64 cols). Partials (o, m, l) combined in LDS.
// ---------------------------------------------------------------------------
__global__ __launch_bounds__(256) void attn_kernel(
    const float* __restrict__ q, const float* __restrict__ k,
    const float* __restrict__ v, float* __restrict__ o) {
  __shared__ float smem[8][1024];  // per wave: P stage (16x64) / partial publish
  const int tid = threadIdx.x;
  const int wave = tid >> 5, lane = tid & 31;
  const int m = lane & 15, half = lane >> 4, kOff = half * 2;
  const int tile = wave >> 2, split = wave & 3;
  const int task = blockIdx.x * 2 + tile;  // 1024 row tiles total
  const int b = task >> 8;
  const int i0 = (task & 255) * 16;
  const float* qb = q + b * 4096 * 32;
  const float* kb = k + b * 4096 * 32;
  const float* vb = v + b * 4096 * 32;

  v2f aK[8];  // k rows as A tiles, D=32 -> 8 K-steps of 4
#pragma unroll
  for (int t = 0; t < 8; ++t) aK[t] = ld2(&kb[(i0 + m) * 32 + 4 * t + kOff]);

  float mrun[8], lrun[8];
  v8f o0 = {}, o1 = {};
#pragma unroll
  for (int vi = 0; vi < 8; ++vi) { mrun[vi] = -__builtin_inff(); lrun[vi] = 0.f; }

  const int jbeg = split * 1024, jend = jbeg + 1024;
  for (int j0 = jbeg; j0 < jend; j0 += 64) {
    // ---- S block: 4 sub-tiles of 16 cols, K(16x32) x Q^T(32x16) each
    v8f s[4] = {{}, {}, {}, {}};
#pragma unroll
    for (int st = 0; st < 4; ++st) {
#pragma unroll
      for (int t = 0; t < 8; ++t) {
        const v2f bq = ld2(&qb[(j0 + st * 16 + m) * 32 + 4 * t + kOff]);
        s[st] = WMMA_F32(aK[t], bq, s[st]);
      }
    }
    // ---- online softmax over the 64-col block (one butterfly set per row)
#pragma unroll
    for (int vi = 0; vi < 8; ++vi) {
      float mx = fmaxf(fmaxf(s[0][vi], s[1][vi]), fmaxf(s[2][vi], s[3][vi]));
      mx = fmaxf(mx, __shfl_xor(mx, 1, 16));
      mx = fmaxf(mx, __shfl_xor(mx, 2, 16));
      mx = fmaxf(mx, __shfl_xor(mx, 4, 16));
      mx = fmaxf(mx, __shfl_xor(mx, 8, 16));
      const float nm = fmaxf(mrun[vi], mx);
      const float sc = __expf(mrun[vi] - nm);
      mrun[vi] = nm;
      const float p0 = __expf(s[0][vi] - nm); s[0][vi] = p0;
      const float p1 = __expf(s[1][vi] - nm); s[1][vi] = p1;
      const float p2 = __expf(s[2][vi] - nm); s[2][vi] = p2;
      const float p3 = __expf(s[3][vi] - nm); s[3][vi] = p3;
      float rs = (p0 + p1) + (p2 + p3);
      rs += __shfl_xor(rs, 1, 16);
      rs += __shfl_xor(rs, 2, 16);
      rs += __shfl_xor(rs, 4, 16);
      rs += __shfl_xor(rs, 8, 16);
      lrun[vi] = lrun[vi] * sc + rs;
      o0[vi] *= sc;
      o1[vi] *= sc;
    }
    // ---- stage 16x64 P to LDS (C-layout -> row-major, same-wave DS in-order)
#pragma unroll
    for (int st = 0; st < 4; ++st)
#pragma unroll
      for (int vi = 0; vi < 8; ++vi)
        smem[wave][(vi + half * 8) * 64 + st * 16 + m] = s[st][vi];
    asm volatile("s_wait_dscnt 0" ::: "memory");
    // ---- o += P(16x64) @ V(64x32): 16 K-steps x 2 col tiles
#pragma unroll
    for (int t = 0; t < 16; ++t) {
      const int kk = 4 * t + kOff;
      const v2f aP = *(const v2f*)&smem[wave][m * 64 + kk];
      const v2f bv0 = mk2(vb[(j0 + kk) * 32 + m],      vb[(j0 + kk + 1) * 32 + m]);
      const v2f bv1 = mk2(vb[(j0 + kk) * 32 + 16 + m], vb[(j0 + kk + 1) * 32 + 16 + m]);
      o0 = WMMA_F32(aP, bv0, o0);
      o1 = WMMA_F32(aP, bv1, o1);
    }
  }

  // ---- publish partials (unnormalized o, l, m) to LDS
  asm volatile("s_wait_dscnt 0" ::: "memory");
#pragma unroll
  for (int vi = 0; vi < 8; ++vi) {
    const int r = vi + half * 8;
    smem[wave][r * 32 + m]      = o0[vi];
    smem[wave][r * 32 + 16 + m] = o1[vi];
  }
  if (m == 0) {
#pragma unroll
    for (int vi = 0; vi < 8; ++vi) {
      smem[wave][512 + vi + half * 8] = lrun[vi];  // l partial
      smem[wave][528 + vi + half * 8] = mrun[vi];  // m partial
    }
  }
  __syncthreads();

  // ---- combine the 4 splits (waves 0 and 4 only)
  if (split == 0) {
    const int w0 = wave;
#pragma unroll
    for (int vi = 0; vi < 8; ++vi) {
      const int r = vi + half * 8;
      float mg = smem[w0][528 + r];
      mg = fmaxf(mg, smem[w0 + 1][528 + r]);
      mg = fmaxf(mg, smem[w0 + 2][528 + r]);
      mg = fmaxf(mg, smem[w0 + 3][528 + r]);
      float lg = 0.f, a0 = 0.f, a1 = 0.f;
#pragma unroll
      for (int p = 0; p < 4; ++p) {
        const float wgt = __expf(smem[w0 + p][528 + r] - mg);
        lg += wgt * smem[w0 + p][512 + r];
        a0 += wgt * smem[w0 + p][r * 32 + m];
        a1 += wgt * smem[w0 + p][r * 32 + 16 + m];
      }
      const float inv = 1.f / lg;
      const int g = b * 4096 * 32 + (i0 + r) * 32;
      o[g + m]      = a0 * inv;
      o[g + 16 + m] = a1 * inv;
    }
  }
}

// ---------------------------------------------------------------------------
// Kernel 3: out = gamma * (o @ Wv) + x    (o:[16384,32], Wv:[32,256])
// One wave -> 16 rows x 64 cols (4 col tiles, K=32 in 8 steps).
// ---------------------------------------------------------------------------
__global__ __launch_bounds__(256) void out_kernel(
    const float* __restrict__ o, const float* __restrict__ Wv,
    const float* __restrict__ x, const float* __restrict__ gamma,
    float* __restrict__ out) {
  const int tid = threadIdx.x;
  const int wave = tid >> 5, lane = tid & 31;
  const int m = lane & 15, half = lane >> 4, kOff = half * 2;
  const int task = blockIdx.x * 8 + wave;  // 4096 tasks
  const int row0 = (task >> 2) * 16, col0 = (task & 3) * 64;

  v2f aO[8];
#pragma unroll
  for (int t = 0; t < 8; ++t) aO[t] = ld2(&o[(row0 + m) * 32 + 4 * t + kOff]);

  v8f acc[4] = {{}, {}, {}, {}};
#pragma unroll
  for (int t = 0; t < 8; ++t) {
    const int kk = 4 * t + kOff;
#pragma unroll
    for (int ct = 0; ct < 4; ++ct) {
      const int col = col0 + ct * 16 + m;
      const v2f bw = mk2(Wv[kk * 256 + col], Wv[(kk + 1) * 256 + col]);
      acc[ct] = WMMA_F32(aO[t], bw, acc[ct]);
    }
  }
  const float g = *gamma;
#pragma unroll
  for (int ct = 0; ct < 4; ++ct) {
#pragma unroll
    for (int vi = 0; vi < 8; ++vi) {
      const int idx = (row0 + vi + half * 8) * 256 + col0 + ct * 16 + m;
      out[idx] = g * acc[ct][vi] + x[idx];
    }
  }
}

// ---------------------------------------------------------------------------
extern "C" void kernel_launch(void* const* d_in, const int* in_sizes, int n_in,
                              void* d_out, int out_size, void* d_ws, size_t ws_size,
                              hipStream_t stream) {
  const float* x  = (const float*)d_in[0];
  const float* Wf = (const float*)d_in[1];
  const float* Wg = (const float*)d_in[2];
  const float* Wh = (const float*)d_in[3];
  const float* Wv = (const float*)d_in[4];
  const float* gm = (const float*)d_in[5];
  float* out = (float*)d_out;

  const int QKV = 4 * 4096 * 32;  // 524288 floats per buffer
  float* q = (float*)d_ws;        // workspace: 4 buffers * 2 MB = 8 MB
  float* k = q + QKV;
  float* v = k + QKV;
  float* o = v + QKV;

  qkv_kernel<<<128, 256, 0, stream>>>(x, Wf, Wg, Wh, q, k, v);
  attn_kernel<<<512, 256, 0, stream>>>(q, k, v, o);
  out_kernel<<<512, 256, 0, stream>>>(o, Wv, x, gm, out);
}